// SpatialGate_28767690948728
// MI455X (gfx1250) — compile-verified
//
#include <hip/hip_runtime.h>
#include <hip/hip_bf16.h>
#include <math.h>

typedef __attribute__((ext_vector_type(2))) float v2f;
typedef __attribute__((ext_vector_type(4))) float v4f;
typedef __attribute__((ext_vector_type(8))) float v8f;

#define HW   3136      // 56*56
#define Wd   56
#define Cch  512
#define Bn   32
#define KD   768

// ---------------------------------------------------------------------------
// Pass 1: channel max + mean pooling.  x:[B,C,H,W] -> comp:[B,2,HW]
// 2 pixels per thread, float2 (b64) loads, coalesced within each channel plane.
// ---------------------------------------------------------------------------
__global__ void pool_kernel(const float* __restrict__ x, float* __restrict__ comp) {
    int idx = blockIdx.x * blockDim.x + threadIdx.x;   // 0 .. B*HW/2-1 (exact)
    int b   = idx / (HW / 2);
    int sp2 = idx - b * (HW / 2);
    const float2* xb = (const float2*)(x + (size_t)b * Cch * HW) + sp2;
    float mx0 = -INFINITY, mx1 = -INFINITY;
    float s0 = 0.0f, s1 = 0.0f;
#pragma unroll 8
    for (int c = 0; c < Cch; ++c) {
        float2 v = xb[c * (HW / 2)];
        mx0 = fmaxf(mx0, v.x);
        mx1 = fmaxf(mx1, v.y);
        s0 += v.x;
        s1 += v.y;
    }
    float2* cb = (float2*)(comp + (size_t)b * 2 * HW) + sp2;
    cb[0]      = make_float2(mx0, mx1);
    cb[HW / 2] = make_float2(s0 * (1.0f / (float)Cch), s1 * (1.0f / (float)Cch));
}

// ---------------------------------------------------------------------------
// Keyword bias: kbias[b] = keyword[b,:] . proj_w + proj_b   (32 x 768 dots)
// ---------------------------------------------------------------------------
__global__ void kbias_kernel(const float* __restrict__ keyword,
                             const float* __restrict__ proj_w,
                             const float* __restrict__ proj_b,
                             float* __restrict__ kbias) {
    __shared__ float red[256];
    int b = blockIdx.x;
    int t = threadIdx.x;
    float s = 0.0f;
    for (int i = t; i < KD; i += 256) s += keyword[b * KD + i] * proj_w[i];
    red[t] = s;
    __syncthreads();
    for (int off = 128; off > 0; off >>= 1) {
        if (t < off) red[t] += red[t + off];
        __syncthreads();
    }
    if (t == 0) kbias[b] = red[0] + proj_b[0];
}

// ---------------------------------------------------------------------------
// Pass 2: 7x7 conv (2->1 ch, SAME pad) + conv_b + kbias, sigmoid -> scale[B,HW]
// V_WMMA_F32_16X16X4_F32, wave32. One wave computes 16 consecutive pixels.
//   A (16x4): lane L<16 = row M=L with K=k0,k0+1 in (a.x,a.y);
//             lane L>=16 = row M=L-16 with K=k0+2,k0+3.
//   B (4x16): weight w[k] replicated across N columns -> every D column is the
//             98-length dot; lane 0 holds M=0..7, lane 16 holds M=8..15.
// LDS table tbl[k] = { bits(w[k]), dh | dw<<8 | ic<<16 }; pad entries (k>=98)
// use dh=127 so they always fail the bounds test (and weight 0).
// Hot loop is fully unrolled and branchless (clamped gather + select).
// ---------------------------------------------------------------------------
__global__ void conv_sigmoid_wmma(const float* __restrict__ comp,   // [B,2,HW]
                                  const float* __restrict__ conv_w, // 98
                                  const float* __restrict__ conv_b, // 1
                                  const float* __restrict__ kbias,  // [B]
                                  float* __restrict__ scale)        // [B,HW]
{
    __shared__ int2 tbl[100];
    int t = threadIdx.x;
    if (t < 100) {
        float wv = (t < 98) ? conv_w[t] : 0.0f;
        int ic = t / 49;
        int r  = t - ic * 49;
        int dh = r / 7;
        int dw = r - dh * 7;
        if (t >= 98) { dh = 127; dw = 127; ic = 0; }
        tbl[t] = make_int2(__float_as_int(wv), dh | (dw << 8) | (ic << 16));
    }
    __syncthreads();

    int wave  = (blockIdx.x * blockDim.x + t) >> 5;   // 8 waves/block, exact grid
    int lane  = t & 31;
    int b     = wave / 196;
    int chunk = wave - b * 196;
    int pbase = chunk * 16;

    int lpix  = lane & 15;
    int p     = pbase + lpix;
    int h     = p / Wd;
    int w     = p - h * Wd;
    int hm3   = h - 3;
    int wm3   = w - 3;
    int khalf = (lane >> 4) * 2;    // lanes 16..31 own K+2,K+3

    const float* cbase = comp + (size_t)b * 2 * HW;

    v8f acc = {};
#pragma unroll
    for (int kt = 0; kt < 25; ++kt) {
        int kb = kt * 4 + khalf;                  // even -> tbl[kb] is 16B aligned
        int4 ee = *(const int4*)(&tbl[kb]);       // one ds_load_b128: k, k+1
        v2f a, bm;
#pragma unroll
        for (int j = 0; j < 2; ++j) {
            int wbits = (j == 0) ? ee.x : ee.z;
            int geom  = (j == 0) ? ee.y : ee.w;
            bm[j] = __int_as_float(wbits);
            int dh = geom & 0xff;
            int dw = (geom >> 8) & 0xff;
            int ic = geom >> 16;
            int hh = hm3 + dh;
            int ww = wm3 + dw;
            bool inb = ((unsigned)hh < (unsigned)Wd) & ((unsigned)ww < (unsigned)Wd);
            int hc = min(max(hh, 0), Wd - 1);
            int wc = min(max(ww, 0), Wd - 1);
            float v = cbase[ic * HW + hc * Wd + wc];   // always in-bounds
            a[j] = inb ? v : 0.0f;
        }
        acc = __builtin_amdgcn_wmma_f32_16x16x4_f32(
                  false, a, false, bm, (short)0, acc, false, false);
    }

    float bias = conv_b[0] + kbias[b];
    if (lane == 0 || lane == 16) {
        int moff = (lane == 16) ? 8 : 0;
        float* out = scale + (size_t)b * HW + pbase + moff;
#pragma unroll
        for (int j = 0; j < 8; ++j) {
            float v = acc[j] + bias;
            out[j] = 1.0f / (1.0f + __expf(-v));
        }
    }
}

// ---------------------------------------------------------------------------
// Pass 3: out[b,c,p] = x[b,c,p] * scale[b,p].
// Dominant traffic (410 MB): x and out are pure streams with no reuse at this
// point -> non-temporal b128 loads/stores; scale (0.4 MB, reused 512x) stays RT.
// ---------------------------------------------------------------------------
__global__ void scale_kernel(const float* __restrict__ x,
                             const float* __restrict__ scale,
                             float* __restrict__ out) {
    unsigned int i = (blockIdx.x * blockDim.x + threadIdx.x) * 4u;  // elem index
    unsigned int b  = i / (unsigned int)(Cch * HW);
    unsigned int sp = i % (unsigned int)HW;
    v4f xv = __builtin_nontemporal_load((const v4f*)(x + i));
    const float4 sv = *(const float4*)(scale + (size_t)b * HW + sp);
    v4f o;
    o.x = xv.x * sv.x;
    o.y = xv.y * sv.y;
    o.z = xv.z * sv.z;
    o.w = xv.w * sv.w;
    __builtin_nontemporal_store(o, (v4f*)(out + i));
}

// ---------------------------------------------------------------------------
extern "C" void kernel_launch(void* const* d_in, const int* in_sizes, int n_in,
                              void* d_out, int out_size, void* d_ws, size_t ws_size,
                              hipStream_t stream) {
    const float* x       = (const float*)d_in[0];  // [32,512,56,56]
    const float* keyword = (const float*)d_in[1];  // [32,768]
    const float* conv_w  = (const float*)d_in[2];  // [1,2,7,7] = 98
    const float* conv_b  = (const float*)d_in[3];  // [1]
    const float* proj_w  = (const float*)d_in[4];  // [1,768]
    const float* proj_b  = (const float*)d_in[5];  // [1]
    float* out = (float*)d_out;

    // scratch layout (1.2 MB total)
    char* ws = (char*)d_ws;
    float* comp  = (float*)(ws);                       // 32*2*3136*4 = 802816 B
    float* scale = (float*)(ws + 802816);              // 32*3136*4   = 401408 B
    float* kbias = (float*)(ws + 802816 + 401408);     // 32*4        = 128 B

    // Pass 1: pooling. 32*3136/2 = 50176 threads -> 196 blocks of 256 (exact)
    pool_kernel<<<196, 256, 0, stream>>>(x, comp);

    // keyword bias: one block per batch
    kbias_kernel<<<Bn, 256, 0, stream>>>(keyword, proj_w, proj_b, kbias);

    // Pass 2: WMMA conv + sigmoid. 32*196 = 6272 waves, 8 waves/block -> 784
    conv_sigmoid_wmma<<<784, 256, 0, stream>>>(comp, conv_w, conv_b, kbias, scale);

    // Pass 3: elementwise scale. 51380224/4 = 12845056 threads -> 50176 blocks
    scale_kernel<<<50176, 256, 0, stream>>>(x, scale, out);
}